// Decoder_50337016709404
// MI455X (gfx1250) — compile-verified
//
#include <hip/hip_runtime.h>
#include <cmath>

// ---------------- problem constants ----------------
#define BB     64      // batch
#define TKK    1024    // encoder length
#define EE     512     // embed dim
#define CC     512     // conv channels
#define DD     400     // in_dim * r
#define TDEC   64      // decoder steps
#define NLAY   4
#define KCONV  5
#define SQRT_HALF 0.70710678118654752440f

typedef __attribute__((ext_vector_type(2))) float v2f;
typedef __attribute__((ext_vector_type(4))) float v4f;
typedef __attribute__((ext_vector_type(8))) float v8f;

__device__ __forceinline__ v8f wmma4(v2f a, v2f b, v8f c) {
  // D = A(16x4 f32) * B(4x16 f32) + C(16x16 f32)
  return __builtin_amdgcn_wmma_f32_16x16x4_f32(false, a, false, b, (short)0, c, false, false);
}

__device__ __forceinline__ float sigmoidf_(float x) { return 1.0f / (1.0f + __expf(-x)); }

// DeepVoice3 sinusoid for decoder frame positions (rate 1.0, dim 512, pos=t+1>0)
__device__ __forceinline__ float fpe_val(int c, int t) {
  float ex  = (float)(2 * (c >> 1)) * (1.0f / 512.0f);
  float div = __powf(10000.0f, ex);
  float ang = (float)(t + 1) / div;
  return (c & 1) ? __cosf(ang) : __sinf(ang);
}

// sinusoid for key positions (rate 1.29, dim 512); pos==0 -> 0
__device__ __forceinline__ float pe_key(int pos, int e) {
  if (pos == 0) return 0.0f;
  float ex  = (float)(2 * (e >> 1)) * (1.0f / 512.0f);
  float div = __powf(10000.0f, ex);
  float ang = 1.29f * (float)pos / div;
  return (e & 1) ? __cosf(ang) : __sinf(ang);
}

enum { M_PRE = 0, M_GLU_PRE = 1, M_GLU_DEC = 2, M_QPROJ = 3, M_OPROJ = 4, M_LAST = 5 };

// ---------------------------------------------------------------------------
// Generic step GEMM:  hT[o][b] = W[o][k] * X[k][b]  (+ epilogue per MODE)
//   W  : row-major [Mtot, KD] (native param layout; for GLU k = c*5+j equals
//        the flat offset inside a weight row, so A-tile loads are contiguous)
//   X  : activations stored transposed [k][64]  (b contiguous)
//   GLU: B-rows gathered from the 5-slot ring (slot (t+1+j)%5); the block
//        computes the matching 'a' (rows c) and 'g' (rows 512+c) tiles so the
//        gated epilogue is register-local.
//   LDS: A tile [rows][k] (stride 36), B tile TRANSPOSED [n][k] (stride 34) so
//        every WMMA fragment is a single aligned ds_load_b64 — no VALU shuffles.
//   KD/MTOT are compile-time: full chunks stage with zero bound checks
//   (b128 -> b128); the guarded tail chunk exists only when KD%32 != 0.
// block = 256 threads (8 waves), each block -> 64 (o) x 64 (b) outputs.
// ---------------------------------------------------------------------------
template <int MODE, int KD, int MTOT>
__global__ __launch_bounds__(256) void dv3_gemm_step(
    const float* __restrict__ W, const float* __restrict__ bias,
    const float* __restrict__ X, const float* __restrict__ ringL,
    const float* __restrict__ res1, const float* __restrict__ res2,
    float* __restrict__ dst, float* __restrict__ dst2, float* __restrict__ dst3,
    int t)
{
  constexpr bool IS_GLU = (MODE == M_GLU_PRE || MODE == M_GLU_DEC);
  constexpr int ROWS_A = IS_GLU ? 128 : 64;
  constexpr int AST = 36;   // A row stride: 16 rows -> 16 distinct banks, f2-aligned
  constexpr int BST = 34;   // B^T row stride: 16 rows -> 16 distinct banks, f2-aligned
  constexpr int NFULL = KD >> 5;
  constexpr int TAILK = KD & 31;
  __shared__ float As[ROWS_A * AST];
  __shared__ float Bs[64 * BST];          // transposed: Bs[n*BST + k]

  const int tid  = threadIdx.x;
  const int m0   = blockIdx.x * 64;
  const int wv   = tid >> 5;
  const int lane = tid & 31;
  const int lm   = lane & 15;
  const int lk   = (lane >> 4) << 1;   // 0 or 2
  const int mi   = wv >> 1;            // m-tile 0..3
  const int nt0  = (wv & 1) << 1;      // first of two n-tiles

  v8f acc0  = {0,0,0,0,0,0,0,0};
  v8f acc1  = {0,0,0,0,0,0,0,0};
  v8f accG0 = {0,0,0,0,0,0,0,0};
  v8f accG1 = {0,0,0,0,0,0,0,0};

  // per-wave LDS fragment base addresses (k-step offsets become immediates)
  const float* aRow  = &As[(mi * 16 + lm) * AST + lk];
  const float* gRow  = &As[(64 + mi * 16 + lm) * AST + lk];
  const float* b0Row = &Bs[(nt0 * 16 + lm) * BST + lk];
  const float* b1Row = &Bs[((nt0 + 1) * 16 + lm) * BST + lk];

  // staging thread mapping (constant across chunks)
  const int ar   = tid >> 3;           // A: row within 32-row group
  const int aq   = (tid & 7) << 2;     // A: 4-float segment within k-chunk
  const int br   = tid >> 3;           // B: local k row 0..31
  const int bseg = (tid & 7) << 3;     // B: 8 consecutive b per thread

  // shared compute body: 8 WMMA k-steps over a staged chunk
  auto compute = [&]() {
#pragma unroll
    for (int kk = 0; kk < 32; kk += 4) {
      v2f a  = *(const v2f*)(aRow + kk);
      v2f b0 = *(const v2f*)(b0Row + kk);
      v2f b1 = *(const v2f*)(b1Row + kk);
      acc0 = wmma4(a, b0, acc0);
      acc1 = wmma4(a, b1, acc1);
      if constexpr (IS_GLU) {
        v2f g = *(const v2f*)(gRow + kk);
        accG0 = wmma4(g, b0, accG0);
        accG1 = wmma4(g, b1, accG1);
      }
    }
  };

  // -------- full chunks: no bound checks --------
  for (int ch = 0; ch < NFULL; ++ch) {
    const int kbase = ch << 5;
    __syncthreads();
    // stage A (b128 -> b128); only M_LAST needs a row guard (OOB protection)
#pragma unroll
    for (int base = 0; base < ROWS_A; base += 32) {
      const int r = base + ar;
      int wrow;
      if constexpr (IS_GLU) wrow = (r < 64) ? (m0 + r) : (512 + m0 + (r - 64));
      else wrow = m0 + r;
      const float* src = W + (size_t)wrow * KD + kbase + aq;
      if constexpr (MTOT % 64 != 0) {
        v4f z = {0, 0, 0, 0};
        *(v4f*)&As[r * AST + aq] = (wrow < MTOT) ? *(const v4f*)src : z;
      } else {
        *(v4f*)&As[r * AST + aq] = *(const v4f*)src;
      }
    }
    // stage B transposed: global row (b contiguous) -> Bs[n][k]
    {
      const int k = kbase + br;
      const float* row;
      if constexpr (IS_GLU) {
        const int j = k % KCONV, c = k / KCONV;
        const int slot = (t + 1 + j) % KCONV;
        row = ringL + ((size_t)slot * CC + c) * BB;
      } else {
        row = X + (size_t)k * BB;
      }
      v4f u0 = *(const v4f*)(row + bseg);
      v4f u1 = *(const v4f*)(row + bseg + 4);
      Bs[(bseg + 0) * BST + br] = u0.x;
      Bs[(bseg + 1) * BST + br] = u0.y;
      Bs[(bseg + 2) * BST + br] = u0.z;
      Bs[(bseg + 3) * BST + br] = u0.w;
      Bs[(bseg + 4) * BST + br] = u1.x;
      Bs[(bseg + 5) * BST + br] = u1.y;
      Bs[(bseg + 6) * BST + br] = u1.z;
      Bs[(bseg + 7) * BST + br] = u1.w;
    }
    __syncthreads();
    compute();
  }

  // -------- peeled tail chunk (only when KD % 32 != 0; e.g. pre: KD=400) ----
  if constexpr (TAILK != 0) {
    const int kbase = NFULL << 5;
    __syncthreads();
#pragma unroll
    for (int base = 0; base < ROWS_A; base += 32) {
      const int r = base + ar;
      int wrow;
      if constexpr (IS_GLU) wrow = (r < 64) ? (m0 + r) : (512 + m0 + (r - 64));
      else wrow = m0 + r;
      const bool rowok = (MTOT % 64 == 0) || (wrow < MTOT);
      const float* src = W + (size_t)wrow * KD + kbase + aq;
#pragma unroll
      for (int i = 0; i < 4; ++i) {
        const int k = kbase + aq + i;
        As[r * AST + aq + i] = (rowok && k < KD) ? src[i] : 0.0f;
      }
    }
    {
      const int k = kbase + br;
      if (k < KD) {
        const float* row;
        if constexpr (IS_GLU) {
          const int j = k % KCONV, c = k / KCONV;
          const int slot = (t + 1 + j) % KCONV;
          row = ringL + ((size_t)slot * CC + c) * BB;
        } else {
          row = X + (size_t)k * BB;
        }
        v4f u0 = *(const v4f*)(row + bseg);
        v4f u1 = *(const v4f*)(row + bseg + 4);
        Bs[(bseg + 0) * BST + br] = u0.x;
        Bs[(bseg + 1) * BST + br] = u0.y;
        Bs[(bseg + 2) * BST + br] = u0.z;
        Bs[(bseg + 3) * BST + br] = u0.w;
        Bs[(bseg + 4) * BST + br] = u1.x;
        Bs[(bseg + 5) * BST + br] = u1.y;
        Bs[(bseg + 6) * BST + br] = u1.z;
        Bs[(bseg + 7) * BST + br] = u1.w;
      } else {
#pragma unroll
        for (int i = 0; i < 8; ++i)
          Bs[(bseg + i) * BST + br] = 0.0f;
      }
    }
    __syncthreads();
    compute();
  }

  // ---- epilogue: C/D layout -> element (m = r + 8*(lane>=16), n = lane&15) ----
#pragma unroll
  for (int r = 0; r < 8; ++r) {
    const int mloc = r + ((lane >> 4) << 3);
    const int o = m0 + mi * 16 + mloc;
#pragma unroll
    for (int q = 0; q < 2; ++q) {
      const int bb = (nt0 + q) * 16 + lm;
      float v  = q ? acc1[r]  : acc0[r];
      float vg = q ? accG1[r] : accG0[r];
      if constexpr (MODE == M_PRE) {
        v += bias[o];
        v = fmaxf(v, 0.0f);
        dst[o * BB + bb] = v;
      } else if constexpr (IS_GLU) {
        float va = v + bias[o];
        float g2 = vg + bias[512 + o];
        float x = va * sigmoidf_(g2);
        if constexpr (MODE == M_GLU_PRE) {
          x = (x + res1[o * BB + bb]) * SQRT_HALF;   // residual = layer input
        } else {
          x += fpe_val(o, t);                        // res2 = x + frame PE
        }
        dst[o * BB + bb] = x;
      } else if constexpr (MODE == M_QPROJ) {
        dst[o * BB + bb] = v + bias[o];
      } else if constexpr (MODE == M_OPROJ) {
        v += bias[o];
        v = (v + res1[o * BB + bb]) * SQRT_HALF;     // + res2 (post-PE x)
        v = (v + res2[o * BB + bb]) * SQRT_HALF;     // + decoder-layer residual
        dst[o * BB + bb] = v;
        if (dst2) dst2[(size_t)bb * (TDEC * CC) + o] = v;   // states[b,t,c]
      } else {  // M_LAST
        if (o < MTOT) {
          v += bias[o];
          dst[o * BB + bb] = v;                      // pre-sigmoid logits (for done)
          float s = sigmoidf_(v);
          dst2[o * BB + bb] = s;                     // curT for next step
          dst3[(size_t)bb * (TDEC * DD) + o] = s;    // outs[b,t,d]
        }
      }
    }
  }
}

// ---------------------------------------------------------------------------
// Windowed attention: the -inf mask leaves <=4 unmasked key positions, so the
// full [B,1024] score/softmax/context collapses to a tiny window — exact math.
// grid = 64 (one block per batch elem), block = 128 (4 waves, one per window pos)
// ---------------------------------------------------------------------------
__global__ __launch_bounds__(128) void dv3_attn(
    const float* __restrict__ qT, const float* __restrict__ keys,
    const float* __restrict__ values, const int* __restrict__ tpos,
    const int* __restrict__ laCur, int* __restrict__ laNext,
    float* __restrict__ ctxT, float* __restrict__ alignsBase,
    int li, int t, int writeAligns)
{
  __shared__ float sc[4];
  __shared__ float aw[4];
  const int b   = blockIdx.x;
  const int tid = threadIdx.x;
  const int la    = laCur[li];
  const int back  = la - 1;
  const int ahead = la + 3;
  const int lo  = (back > 0) ? back : 0;
  const int hi  = (ahead < TKK) ? ahead : TKK;
  const int win = hi - lo;

  const int wv = tid >> 5, lane = tid & 31;
  if (wv < win) {
    const int p = lo + wv;
    const int pos = tpos[(size_t)b * TKK + p];
    float s = 0.0f;
    for (int e = lane; e < EE; e += 32)
      s += qT[e * BB + b] * (keys[((size_t)b * TKK + p) * EE + e] + pe_key(pos, e));
    for (int off = 16; off; off >>= 1) s += __shfl_down(s, off);
    if (lane == 0) sc[wv] = s;
  }
  __syncthreads();
  if (tid == 0) {
    float m = -1e30f;
    for (int w = 0; w < win; ++w) m = fmaxf(m, sc[w]);
    float sum = 0.0f;
    for (int w = 0; w < win; ++w) { float ev = __expf(sc[w] - m); aw[w] = ev; sum += ev; }
    float inv = 1.0f / sum;
    for (int w = 0; w < win; ++w) aw[w] *= inv;
    if (b == 0) {  // reference: argmax of attn for batch element 0 only
      int best = 0; float bm = aw[0];
      for (int w = 1; w < win; ++w) if (aw[w] > bm) { bm = aw[w]; best = w; }
      laNext[li] = lo + best;
    }
  }
  __syncthreads();
  // ctx = attn @ values * sqrt(TK); sqrt(1024) == 32
  for (int e = tid; e < EE; e += blockDim.x) {
    float acc = 0.0f;
    for (int w = 0; w < win; ++w)
      acc += aw[w] * values[((size_t)b * TKK + lo + w) * EE + e];
    ctxT[e * BB + b] = acc * 32.0f;
  }
  // aligns output = 2 * attn(layer 0)  (faithful to the ave-doubling bug, /NL)
  if (writeAligns) {
    float* row = alignsBase + ((size_t)b * TDEC + t) * TKK;
    for (int p = tid; p < TKK; p += blockDim.x) {
      float v = 0.0f;
      if (p >= lo && p < hi) v = 2.0f * aw[p - lo];
      row[p] = v;
    }
  }
}

// done = sigmoid(o @ fc_w.T + fc_b); also commits laNext -> laCur for next step
__global__ __launch_bounds__(128) void dv3_done(
    const float* __restrict__ oT, const float* __restrict__ fc_w,
    const float* __restrict__ fc_b, int* __restrict__ laCur,
    const int* __restrict__ laNext, float* __restrict__ donesBase, int t)
{
  const int tid = threadIdx.x;
  if (tid < BB) {
    float acc = fc_b[0];
    for (int d = 0; d < DD; ++d) acc += oT[d * BB + tid] * fc_w[d];
    donesBase[(size_t)tid * TDEC + t] = 1.0f / (1.0f + __expf(-acc));
  } else if (tid < BB + NLAY) {
    laCur[tid - BB] = laNext[tid - BB];
  }
}

// ---------------------------------------------------------------------------
extern "C" void kernel_launch(void* const* d_in, const int* in_sizes, int n_in,
                              void* d_out, int out_size, void* d_ws, size_t ws_size,
                              hipStream_t stream) {
  (void)in_sizes; (void)n_in; (void)out_size; (void)ws_size;
  const float* keys   = (const float*)d_in[0];
  const float* values = (const float*)d_in[1];
  const int*   tpos   = (const int*)  d_in[2];
  const float* pre_w  = (const float*)d_in[3];
  const float* pre_b  = (const float*)d_in[4];
  const float* glu_w  = (const float*)d_in[5];   // [8,1024,512,5]
  const float* glu_b  = (const float*)d_in[6];   // [8,1024]
  const float* q_w    = (const float*)d_in[7];   // [4,512,512]
  const float* q_b    = (const float*)d_in[8];
  const float* o_w    = (const float*)d_in[9];   // [4,512,512]
  const float* o_b    = (const float*)d_in[10];
  const float* last_w = (const float*)d_in[11];  // [400,512]
  const float* last_b = (const float*)d_in[12];
  const float* fc_w   = (const float*)d_in[13];  // [1,400]
  const float* fc_b   = (const float*)d_in[14];
  // d_in[15] = num_steps (== TDEC, compile-time here)

  float* out = (float*)d_out;
  float* outs   = out;
  float* aligns = out + (size_t)BB * TDEC * DD;
  float* dones  = aligns + (size_t)BB * TDEC * TKK;
  float* states = dones + (size_t)BB * TDEC;

  // workspace carve (~6 MB): ring buffers + transposed activations
  float* ring  = (float*)d_ws;                       // [8][5][512][64]
  float* curT  = ring + (size_t)8 * KCONV * CC * BB; // [400][64]
  int*   laCur = (int*)(curT + (size_t)DD * BB);
  int*   laNext = laCur + NLAY;
  float* res2T = (float*)(laNext + NLAY);            // [512][64]
  float* qT    = res2T + (size_t)CC * BB;
  float* ctxT  = qT + (size_t)EE * BB;
  float* xfinT = ctxT + (size_t)EE * BB;
  float* oT    = xfinT + (size_t)CC * BB;            // [400][64] pre-sigmoid

  // zero conv ring state, initial cur, and last_attended (both buffers)
  const size_t zeroBytes =
      ((size_t)8 * KCONV * CC * BB + (size_t)DD * BB) * sizeof(float) +
      2 * NLAY * sizeof(int);
  hipMemsetAsync(d_ws, 0, zeroBytes, stream);

  auto ringSlot = [&](int L, int slot) {
    return ring + (((size_t)L * KCONV + slot) * CC) * BB;
  };

  for (int t = 0; t < TDEC; ++t) {
    const int slot = t % KCONV;
    // pre: x = relu(cur @ pre_w.T + pre_b) -> ring[0][slot]
    dv3_gemm_step<M_PRE, DD, CC><<<dim3(CC / 64), 256, 0, stream>>>(
        pre_w, pre_b, curT, nullptr, nullptr, nullptr,
        ringSlot(0, slot), nullptr, nullptr, t);

    // preattention Conv1dGLU stack (residual=True)
    for (int L = 0; L < NLAY; ++L) {
      dv3_gemm_step<M_GLU_PRE, KCONV * CC, CC><<<dim3(CC / 64), 256, 0, stream>>>(
          glu_w + (size_t)L * 1024 * 2560, glu_b + (size_t)L * 1024,
          nullptr, ring + (size_t)L * KCONV * CC * BB,
          ringSlot(L, slot), nullptr,
          ringSlot(L + 1, slot), nullptr, nullptr, t);
    }

    // decoder Conv1dGLU + attention blocks
    for (int li = 0; li < NLAY; ++li) {
      const int L = NLAY + li;
      // GLU (residual=False) + frame PE -> res2T
      dv3_gemm_step<M_GLU_DEC, KCONV * CC, CC><<<dim3(CC / 64), 256, 0, stream>>>(
          glu_w + (size_t)L * 1024 * 2560, glu_b + (size_t)L * 1024,
          nullptr, ring + (size_t)L * KCONV * CC * BB,
          nullptr, nullptr, res2T, nullptr, nullptr, t);
      // q projection
      dv3_gemm_step<M_QPROJ, CC, EE><<<dim3(EE / 64), 256, 0, stream>>>(
          q_w + (size_t)li * EE * CC, q_b + (size_t)li * EE,
          res2T, nullptr, nullptr, nullptr, qT, nullptr, nullptr, t);
      // windowed attention -> ctxT (+ aligns for layer 0, + argmax -> laNext)
      dv3_attn<<<dim3(BB), 128, 0, stream>>>(
          qT, keys, values, tpos, laCur, laNext, ctxT, aligns, li, t,
          (li == 0) ? 1 : 0);
      // out projection + both residual combines
      float* odst = (li < 3) ? ringSlot(NLAY + li + 1, slot) : xfinT;
      float* sdst = (li == 3) ? (states + (size_t)t * CC) : nullptr;
      dv3_gemm_step<M_OPROJ, EE, CC><<<dim3(CC / 64), 256, 0, stream>>>(
          o_w + (size_t)li * CC * EE, o_b + (size_t)li * CC,
          ctxT, nullptr, res2T, ringSlot(L, slot),
          odst, sdst, nullptr, t);
    }

    // last_conv 512->400, sigmoid -> outs + curT; pre-sigmoid -> oT
    dv3_gemm_step<M_LAST, CC, DD><<<dim3((DD + 63) / 64), 256, 0, stream>>>(
        last_w, last_b, xfinT, nullptr, nullptr, nullptr,
        oT, curT, outs + (size_t)t * DD, t);

    // done flag + commit last_attended for next step
    dv3_done<<<dim3(1), 128, 0, stream>>>(oT, fc_w, fc_b, laCur, laNext, dones, t);
  }
}